// OptimalAlignmentLoss_35407710388919
// MI455X (gfx1250) — compile-verified
//
#include <hip/hip_runtime.h>

#define EPSF 1e-6f
#define OUTF 10.0f
#define NSEL 256
#define MCOL 768          // 3*NSEL
#define KCAND 128
#define PIX (4*512*512)   // B*H*W
#define NPART 512         // partial-reduction blocks

typedef float v2f __attribute__((ext_vector_type(2)));
typedef float v8f __attribute__((ext_vector_type(8)));

// ---------- helpers ----------
__device__ __forceinline__ unsigned orderF(float f) {
  unsigned u = __float_as_uint(f);
  return (u & 0x80000000u) ? ~u : (u | 0x80000000u);
}
__device__ __forceinline__ float unorderF(unsigned u) {
  unsigned v = (u & 0x80000000u) ? (u & 0x7FFFFFFFu) : ~u;
  return __uint_as_float(v);
}
__device__ __forceinline__ unsigned rotl32(unsigned v, int n) { return (v << n) | (v >> (32 - n)); }

// JAX threefry2x32 (16 rounds), key = (0,42) for jax.random.key(42)
__device__ void threefry2x32(unsigned k0, unsigned k1, unsigned c0, unsigned c1,
                             unsigned* o0, unsigned* o1) {
  unsigned ks2 = 0x1BD11BDAu ^ k0 ^ k1;
  unsigned x0 = c0 + k0, x1 = c1 + k1;
  x0 += x1; x1 = rotl32(x1, 13); x1 ^= x0;
  x0 += x1; x1 = rotl32(x1, 15); x1 ^= x0;
  x0 += x1; x1 = rotl32(x1, 26); x1 ^= x0;
  x0 += x1; x1 = rotl32(x1, 6);  x1 ^= x0;
  x0 += k1; x1 += ks2 + 1u;
  x0 += x1; x1 = rotl32(x1, 17); x1 ^= x0;
  x0 += x1; x1 = rotl32(x1, 29); x1 ^= x0;
  x0 += x1; x1 = rotl32(x1, 16); x1 ^= x0;
  x0 += x1; x1 = rotl32(x1, 24); x1 ^= x0;
  x0 += ks2; x1 += k0 + 2u;
  x0 += x1; x1 = rotl32(x1, 13); x1 ^= x0;
  x0 += x1; x1 = rotl32(x1, 15); x1 ^= x0;
  x0 += x1; x1 = rotl32(x1, 26); x1 ^= x0;
  x0 += x1; x1 = rotl32(x1, 6);  x1 ^= x0;
  x0 += k0; x1 += k1 + 3u;
  x0 += x1; x1 = rotl32(x1, 17); x1 ^= x0;
  x0 += x1; x1 = rotl32(x1, 29); x1 ^= x0;
  x0 += x1; x1 = rotl32(x1, 16); x1 ^= x0;
  x0 += x1; x1 = rotl32(x1, 24); x1 ^= x0;
  x0 += k1; x1 += ks2 + 4u;
  *o0 = x0; *o1 = x1;
}

__device__ void bitonic_u64(unsigned long long* a, int n, int tid, int nt) {
  for (int k = 2; k <= n; k <<= 1) {
    for (int j = k >> 1; j > 0; j >>= 1) {
      for (int i = tid; i < n; i += nt) {
        int ixj = i ^ j;
        if (ixj > i) {
          unsigned long long x = a[i], y = a[ixj];
          bool asc = (i & k) == 0;
          bool sw = asc ? (x > y) : (x < y);
          if (sw) { a[i] = y; a[ixj] = x; }
        }
      }
      __syncthreads();
    }
  }
}
// searchsorted: left = count of (<v), right = count of (<=v)
__device__ __forceinline__ int lowerB(const float* a, int n, float v) {
  int lo = 0, hi = n;
  while (lo < hi) { int m = (lo + hi) >> 1; if (a[m] < v) lo = m + 1; else hi = m; }
  return lo;
}
__device__ __forceinline__ int upperB(const float* a, int n, float v) {
  int lo = 0, hi = n;
  while (lo < hi) { int m = (lo + hi) >> 1; if (a[m] <= v) lo = m + 1; else hi = m; }
  return lo;
}

// ---------- K1: per-batch RNG + top_k selection + gather P,Q ----------
__global__ __launch_bounds__(1024) void kSelect(const float* __restrict__ ptsR,
                                                const float* __restrict__ texR,
                                                const float* __restrict__ outP,
                                                const unsigned char* __restrict__ masks,
                                                float* selP, float* selQ, int* hasF, int b) {
  __shared__ unsigned long long keys[4096];
  int tid = threadIdx.x;
  for (int t = tid; t < 4096; t += 1024) {
    int py = t >> 6, px = t & 63;
    int y = py * 8, x = px * 8;
    unsigned g = (unsigned)(b * 4096 + t);
    unsigned o0, o1, bits;
    if (g < 8192u) { threefry2x32(0u, 42u, g, g + 8192u, &o0, &o1); bits = o0; }
    else           { threefry2x32(0u, 42u, g - 8192u, g, &o0, &o1); bits = o1; }
    float r = __uint_as_float((bits >> 9) | 0x3F800000u) - 1.0f;
    bool mk = masks[(b * 512 + y) * 512 + x] != 0;
    float score = mk ? r : -1.0f;
    unsigned inv = ~orderF(score);  // descending score, ascending index on ties
    keys[t] = ((unsigned long long)inv << 32) | (unsigned long long)t;
  }
  __syncthreads();
  bitonic_u64(keys, 4096, tid, 1024);
  if (tid < NSEL) {
    int p = (int)(keys[tid] & 0xFFFFFFFFull);
    int y = (p >> 6) * 8, x = (p & 63) * 8;
    long long pix = ((long long)b * 512 + y) * 512 + x;
    // stable min of (valid ? z : inf) over the 4 layers  == sorted layer 0
    float bestz = __builtin_inff(); int lmin = 0;
    for (int l = 0; l < 4; l++) {
      float a = texR[(pix * 4 + l) * 4 + 3];
      float z = ptsR[(pix * 4 + l) * 3 + 2];
      float zz = (a > 0.9f) ? z : __builtin_inff();
      if (zz < bestz) { bestz = zz; lmin = l; }
    }
    for (int c = 0; c < 3; c++) {
      selP[(b * NSEL + tid) * 3 + c] = ptsR[(pix * 4 + lmin) * 3 + c];
      selQ[(b * NSEL + tid) * 3 + c] = outP[(pix * 4 + 0) * 3 + c];
    }
  }
  if (tid == 0) hasF[b] = 0;
}

// ---------- K2: per-row (256 blocks) build+sort+cumsum+grad -> extrema, ratio ----------
__global__ __launch_bounds__(256) void kSolveRows(const float* __restrict__ selP,
                                                  const float* __restrict__ selQ,
                                                  float* ratioG, unsigned char* exG,
                                                  int* hasF, int b) {
  __shared__ float sPz[NSEL], sQz[NSEL], sPxy[2 * NSEL], sQxy[2 * NSEL];
  __shared__ float ratio_s[MCOL], rm_s[MCOL], rp_s[MCOL], val_s[MCOL];
  __shared__ unsigned long long sb[1024];
  __shared__ float skey[3][MCOL];
  __shared__ float scum[3][MCOL + 2];
  __shared__ float scanTmp[1024];
  __shared__ int sHas;
  int tid = threadIdx.x, i = blockIdx.x;
  {
    const float* P = &selP[(b * NSEL + tid) * 3];
    const float* Q = &selQ[(b * NSEL + tid) * 3];
    sPxy[2 * tid] = P[0]; sPxy[2 * tid + 1] = P[1]; sPz[tid] = P[2];
    sQxy[2 * tid] = Q[0]; sQxy[2 * tid + 1] = Q[1]; sQz[tid] = Q[2];
    if (tid == 0) sHas = 0;
  }
  __syncthreads();
  float Pzi = sPz[i], Qzi = sQz[i];
  for (int m = tid; m < MCOL; m += 256) {
    int j = m / 3, c = m - 3 * j;
    float p2, q2;
    if (c == 0)      { p2 = sPxy[2 * j];     q2 = sQxy[2 * j]; }
    else if (c == 1) { p2 = sPxy[2 * j + 1]; q2 = sQxy[2 * j + 1]; }
    else             { p2 = sPz[j] - Pzi;    q2 = sQz[j] - Qzi; }
    if (!(q2 > 0.f)) { p2 = -p2; q2 = -q2; }
    if (fabsf(q2) < EPSF) { p2 = __builtin_inff(); q2 = EPSF; }
    float w = 1.f / (fabsf(sPz[j]) + EPSF);
    val_s[m] = w * q2;
    float rt = p2 / (q2 + EPSF);
    ratio_s[m] = rt;
    ratioG[i * MCOL + m] = rt;
    rm_s[m] = (w * p2 - OUTF) / (w * q2 + EPSF);
    rp_s[m] = (w * p2 + OUTF) / (w * q2 + EPSF);
  }
  __syncthreads();
  for (int q = 0; q < 3; q++) {
    const float* key = (q == 0) ? ratio_s : ((q == 1) ? rm_s : rp_s);
    for (int t = tid; t < 1024; t += 256)
      sb[t] = (t < MCOL)
        ? (((unsigned long long)orderF(key[t]) << 32) | (unsigned long long)__float_as_uint(val_s[t]))
        : 0xFFFFFFFF00000000ull;   // pad sorts after everything real
    __syncthreads();
    bitonic_u64(sb, 1024, tid, 256);
    for (int t = tid; t < 1024; t += 256) {
      if (t < MCOL) {
        skey[q][t] = unorderF((unsigned)(sb[t] >> 32));
        scanTmp[t] = __uint_as_float((unsigned)sb[t]);
      } else scanTmp[t] = 0.f;
    }
    __syncthreads();
    for (int off = 1; off < 1024; off <<= 1) {   // inclusive scan
      float tadd[4];
      for (int e = 0; e < 4; e++) { int t = tid + e * 256; tadd[e] = (t >= off) ? scanTmp[t - off] : 0.f; }
      __syncthreads();
      for (int e = 0; e < 4; e++) { int t = tid + e * 256; scanTmp[t] += tadd[e]; }
      __syncthreads();
    }
    for (int t = tid; t < MCOL + 2; t += 256)
      scum[q][t] = (t == 0) ? 0.f : ((t <= MCOL) ? scanTmp[t - 1] : __builtin_inff());
    __syncthreads();
  }
  for (int m = tid; m < MCOL; m += 256) {
    float v = ratio_s[m];
    int a0 = lowerB(skey[0], MCOL, v), b0 = upperB(skey[0], MCOL, v);
    int a1 = lowerB(skey[1], MCOL, v), b1 = upperB(skey[1], MCOL, v);
    int a2 = lowerB(skey[2], MCOL, v), b2 = upperB(skey[2], MCOL, v);
    float lg = 2.f * scum[0][a0] - scum[1][a1] - scum[2][a2];
    float rg = 2.f * scum[0][b0] - scum[1][b1] - scum[2][b2];
    bool fin = fabsf(v) < __builtin_inff();
    unsigned char ex = (lg < -EPSF && rg >= EPSF && fin) ? 1 : 0;
    exG[i * MCOL + m] = ex;
    if (ex) sHas = 1;
  }
  __syncthreads();
  if (tid == 0 && sHas) atomicOr(&hasF[b], 1);
}

// ---------- K3: stable top_k of booleans -> 128 candidates ----------
__global__ __launch_bounds__(1024) void kCands(const unsigned char* __restrict__ exG,
                                               const float* __restrict__ ratioG,
                                               const float* __restrict__ selP,
                                               const float* __restrict__ selQ,
                                               int* candIdx, float* candS, float* candT,
                                               int* candOk, int b) {
  __shared__ int wTot[32], wOff[32], sBase, sTot, sN1;
  int tid = threadIdx.x, lane = tid & 31, wid = tid >> 5;
  const int TOTAL = NSEL * MCOL;  // 196608 = 192*1024
  if (tid == 0) sBase = 0;
  __syncthreads();
  for (int pass = 0; pass < 2; pass++) {
    for (int start = 0; start < TOTAL; start += 1024) {
      int idx = start + tid;
      int f = (pass == 0) ? (exG[idx] != 0) : (exG[idx] == 0);
      unsigned mb = (unsigned)__ballot(f);
      if (lane == 0) wTot[wid] = __popc(mb);
      __syncthreads();
      if (tid == 0) { int acc = sBase; for (int w = 0; w < 32; w++) { wOff[w] = acc; acc += wTot[w]; } sTot = acc; }
      __syncthreads();
      if (f) {
        int slot = wOff[wid] + __popc(mb & ((1u << lane) - 1u));
        if (slot < KCAND) candIdx[slot] = idx;
      }
      __syncthreads();
      if (tid == 0) sBase = sTot;
      __syncthreads();
      if (sBase >= KCAND) break;
    }
    if (pass == 0) { if (tid == 0) sN1 = sBase; __syncthreads(); }
  }
  __syncthreads();
  if (tid < KCAND) {
    int idx = candIdx[tid];
    int k = idx / MCOL;
    float s = ratioG[idx];
    if (!(fabsf(s) < __builtin_inff())) s = 0.f;
    float pz = selP[(b * NSEL + k) * 3 + 2];
    float qz = selQ[(b * NSEL + k) * 3 + 2];
    candS[tid] = s;
    candT[tid] = pz - s * qz;
    candOk[tid] = (tid < sN1) ? 1 : 0;
  }
}

// ---------- K4: candidate cost (one block per candidate) ----------
__global__ __launch_bounds__(256) void kCost(const float* __restrict__ selP,
                                             const float* __restrict__ selQ,
                                             const float* __restrict__ candS,
                                             const float* __restrict__ candT,
                                             const int* __restrict__ candOk,
                                             float* costG, int b) {
  __shared__ float red[256];
  int c = blockIdx.x, tid = threadIdx.x;
  float s = candS[c], tz = candT[c];
  const float* P = &selP[(b * NSEL + tid) * 3];
  const float* Q = &selQ[(b * NSEL + tid) * 3];
  float az = fabsf(P[2]);
  float term = (fabsf(P[0] - s * Q[0]) + fabsf(P[1] - s * Q[1]) + fabsf(P[2] - s * Q[2] - tz)) / az;
  red[tid] = term;
  __syncthreads();
  for (int off = 128; off > 0; off >>= 1) { if (tid < off) red[tid] += red[tid + off]; __syncthreads(); }
  if (tid == 0) costG[c] = candOk[c] ? red[0] : __builtin_inff();
}

// ---------- K5: argmin + fallback + final scale/shift per batch ----------
__global__ __launch_bounds__(256) void kFinish(const float* __restrict__ selP,
                                               const float* __restrict__ selQ,
                                               const float* __restrict__ costG,
                                               const float* __restrict__ candS,
                                               const float* __restrict__ candT,
                                               const int* __restrict__ hasF,
                                               float* ssG, int b) {
  __shared__ float sz[NSEL], ipz[NSEL], fc[NSEL];
  int tid = threadIdx.x;
  float pz = selP[(b * NSEL + tid) * 3 + 2];
  float qz = selQ[(b * NSEL + tid) * 3 + 2];
  sz[tid] = pz - qz;
  ipz[tid] = 1.f / fabsf(pz);
  __syncthreads();
  float a = 0.f, szj = sz[tid];
  for (int i = 0; i < NSEL; i++) a += fabsf(sz[i] - szj) * ipz[i];
  fc[tid] = a;
  __syncthreads();
  if (tid == 0) {
    int fb = 0; float bv = fc[0];
    for (int j = 1; j < NSEL; j++) if (fc[j] < bv) { bv = fc[j]; fb = j; }
    int bc = 0; float cv = costG[0];
    for (int c = 1; c < KCAND; c++) if (costG[c] < cv) { cv = costG[c]; bc = c; }
    bool has = hasF[b] != 0;
    ssG[b * 4 + 0] = has ? candS[bc] : 1.0f;
    ssG[b * 4 + 1] = has ? candT[bc] : sz[fb];
    ssG[b * 4 + 2] = 0.f; ssG[b * 4 + 3] = 0.f;
  }
}

// ---------- K6: streaming region loss (bandwidth bound: ~235 MB) ----------
__global__ __launch_bounds__(256) void kRegion(const float* __restrict__ ptsR,
                                               const float* __restrict__ texR,
                                               const float* __restrict__ outP,
                                               const float* __restrict__ outC,
                                               const float* __restrict__ outCf,
                                               const float* __restrict__ ssG,
                                               float* partials) {
  int tid = threadIdx.x;
  int gid = blockIdx.x * 256 + tid;
  float acc[8];
#pragma unroll
  for (int s = 0; s < 8; s++) acc[s] = 0.f;
  const float4* t4  = (const float4*)texR;
  const float4* p4  = (const float4*)ptsR;
  const float4* op4 = (const float4*)outP;
  const float4* oc4 = (const float4*)outC;
  const float4* cf4 = (const float4*)outCf;
  for (int t = gid; t < PIX; t += NPART * 256) {
    int b = t >> 18;
    float scl = ssG[b * 4 + 0], shz = ssG[b * 4 + 1];
    __builtin_prefetch(&t4[(size_t)(t + NPART * 256) * 4], 0, 1);
    float4 a0 = t4[(size_t)t * 4 + 0], a1 = t4[(size_t)t * 4 + 1],
           a2 = t4[(size_t)t * 4 + 2], a3 = t4[(size_t)t * 4 + 3];
    float tex[16] = {a0.x,a0.y,a0.z,a0.w, a1.x,a1.y,a1.z,a1.w,
                     a2.x,a2.y,a2.z,a2.w, a3.x,a3.y,a3.z,a3.w};
    float4 g0 = p4[(size_t)t * 3 + 0], g1 = p4[(size_t)t * 3 + 1], g2 = p4[(size_t)t * 3 + 2];
    float gp[12] = {g0.x,g0.y,g0.z,g0.w, g1.x,g1.y,g1.z,g1.w, g2.x,g2.y,g2.z,g2.w};
    float4 q0 = op4[(size_t)t * 3 + 0], q1 = op4[(size_t)t * 3 + 1], q2 = op4[(size_t)t * 3 + 2];
    float pp[12] = {q0.x,q0.y,q0.z,q0.w, q1.x,q1.y,q1.z,q1.w, q2.x,q2.y,q2.z,q2.w};
    float4 c0 = oc4[(size_t)t * 3 + 0], c1 = oc4[(size_t)t * 3 + 1], c2 = oc4[(size_t)t * 3 + 2];
    float pc[12] = {c0.x,c0.y,c0.z,c0.w, c1.x,c1.y,c1.z,c1.w, c2.x,c2.y,c2.z,c2.w};
    float4 cc = cf4[t];
    float cf[4] = {cc.x, cc.y, cc.z, cc.w};

    float zk[4]; int valv[4];
#pragma unroll
    for (int l = 0; l < 4; l++) {
      int vl = tex[l * 4 + 3] > 0.9f;
      valv[l] = vl;
      zk[l] = vl ? gp[l * 3 + 2] : __builtin_inff();
    }
    int inv[4];
#pragma unroll
    for (int l = 0; l < 4; l++) {
      int r = 0;
#pragma unroll
      for (int m = 0; m < 4; m++)
        r += (zk[m] < zk[l]) || (zk[m] == zk[l] && m < l);  // stable argsort
      inv[r] = l;
    }
#pragma unroll
    for (int r = 0; r < 4; r++) {
      int l = inv[r];
      float v = valv[l] ? 1.f : 0.f;
      float conf = cf[r];
      float gx = gp[l * 3], gy = gp[l * 3 + 1], gz = gp[l * 3 + 2];
      float ax = scl * pp[r * 3]     - gx;
      float ay = scl * pp[r * 3 + 1] - gy;
      float az = scl * pp[r * 3 + 2] + shz - gz;
      float invz = 1.f / (fabsf(gz) + EPSF);
      float d  = v * conf * (fabsf(ax) + fabsf(ay) + fabsf(az)) * invz;
      float cd = v * conf * (fabsf(pc[r * 3]     - tex[l * 4]) +
                             fabsf(pc[r * 3 + 1] - tex[l * 4 + 1]) +
                             fabsf(pc[r * 3 + 2] - tex[l * 4 + 2]));
      float cl = -v * logf(conf + EPSF);
      int s0 = (r == 0) ? 0 : 4;
      acc[s0] += d; acc[s0 + 1] += cd; acc[s0 + 2] += cl; acc[s0 + 3] += v;
    }
  }
  __shared__ float red[256];
  for (int s = 0; s < 8; s++) {
    red[tid] = acc[s];
    __syncthreads();
    for (int off = 128; off > 0; off >>= 1) { if (tid < off) red[tid] += red[tid + off]; __syncthreads(); }
    if (tid == 0) partials[s * NPART + blockIdx.x] = red[0];
    __syncthreads();
  }
}

// ---------- K7: exact-f32 final reduction via V_WMMA_F32_16X16X4_F32 ----------
// B = all-ones (layout-agnostic): D[m][n] += sum_k A[m][k]; total = sum_m D[m][0],
// held in VGPRs 0..7 of lanes 0 (M=0..7) and 16 (M=8..15).
__global__ __launch_bounds__(32) void kFinal(const float* __restrict__ partials, float* out) {
  int lane = threadIdx.x;
  __shared__ float tot[8];
  __shared__ float red[32];
  v2f ones; ones[0] = 1.f; ones[1] = 1.f;
  for (int s = 0; s < 8; s++) {
    v8f c;
#pragma unroll
    for (int k = 0; k < 8; k++) c[k] = 0.f;
    for (int ch = 0; ch < NPART; ch += 64) {
      v2f a;
      a[0] = partials[s * NPART + ch + lane];
      a[1] = partials[s * NPART + ch + 32 + lane];
#if __has_builtin(__builtin_amdgcn_wmma_f32_16x16x4_f32)
      c = __builtin_amdgcn_wmma_f32_16x16x4_f32(false, a, false, ones, (short)0, c,
                                                false, false);
#else
      c[0] += a[0] * ones[0] + a[1] * ones[1];   // fallback (only lanes 0 and 16 are read below: not the full sum)
#endif
    }
    float col = 0.f;
    if (lane == 0 || lane == 16) {
#pragma unroll
      for (int k = 0; k < 8; k++) col += c[k];
    }
    red[lane] = col;
    __syncthreads();
    if (lane == 0) tot[s] = red[0] + red[16];
    __syncthreads();
  }
  if (lane == 0) {
    float dv = tot[3] + 1.f, dox = tot[7] + 1.f;
    out[0] = tot[0] / dv;  out[1] = tot[1] / dv;  out[2] = tot[2] / dv;
    out[3] = tot[4] / dox; out[4] = tot[5] / dox; out[5] = tot[6] / dox;
  }
}

// ---------- launch ----------
extern "C" void kernel_launch(void* const* d_in, const int* in_sizes, int n_in,
                              void* d_out, int out_size, void* d_ws, size_t ws_size,
                              hipStream_t stream) {
  const float* ptsR = (const float*)d_in[0];
  const float* texR = (const float*)d_in[1];
  const unsigned char* masks = (const unsigned char*)d_in[2];
  const float* outP = (const float*)d_in[3];
  const float* outC = (const float*)d_in[4];
  const float* outCf = (const float*)d_in[5];
  float* out = (float*)d_out;
  char* ws = (char*)d_ws;

  float* selP    = (float*)(ws + 0);                 // 4*256*3*4 = 12288
  float* selQ    = (float*)(ws + 12288);             // 12288
  float* ssG     = (float*)(ws + 24576);             // 4*4*4 = 64
  int*   hasF    = (int*)  (ws + 24640);             // 16
  int*   candIdx = (int*)  (ws + 24656);             // 512
  float* candS   = (float*)(ws + 25168);             // 512
  float* candT   = (float*)(ws + 25680);             // 512
  int*   candOk  = (int*)  (ws + 26192);             // 512
  float* costG   = (float*)(ws + 26704);             // 512
  float* partials= (float*)(ws + 27216);             // 8*512*4 = 16384
  float* ratioG  = (float*)(ws + 43600);             // 256*768*4 = 786432
  unsigned char* exG = (unsigned char*)(ws + 830032);// 196608
  (void)in_sizes; (void)n_in; (void)out_size; (void)ws_size;

  for (int b = 0; b < 4; b++) {
    kSelect   <<<1,   1024, 0, stream>>>(ptsR, texR, outP, masks, selP, selQ, hasF, b);
    kSolveRows<<<256, 256,  0, stream>>>(selP, selQ, ratioG, exG, hasF, b);
    kCands    <<<1,   1024, 0, stream>>>(exG, ratioG, selP, selQ, candIdx, candS, candT, candOk, b);
    kCost     <<<128, 256,  0, stream>>>(selP, selQ, candS, candT, candOk, costG, b);
    kFinish   <<<1,   256,  0, stream>>>(selP, selQ, costG, candS, candT, hasF, ssG, b);
  }
  kRegion<<<NPART, 256, 0, stream>>>(ptsR, texR, outP, outC, outCf, ssG, partials);
  kFinal <<<1, 32, 0, stream>>>(partials, out);
}